// WaveformAug_43078521978920
// MI455X (gfx1250) — compile-verified
//
#include <hip/hip_runtime.h>
#include <stdint.h>

#define B_      64
#define T_      320000
#define TILE    2048
#define HW      16          // max k/2 (k <= 33)
#define ELEMS   8           // contiguous outputs per thread
#define THREADS 256
#define CHUNKS  8
#define CHUNK_LEN (T_ / CHUNKS)   // 40000
#define HALO_LEN (TILE + 2 * HW)  // 2080 (multiple of 4)

// clang-native vector (HIP float4 is a struct and rejected by nontemporal builtins)
typedef float v4f __attribute__((ext_vector_type(4)));

// ---------------- CDNA5 async global->LDS staging (ASYNCcnt path) ----------------
__device__ __forceinline__ void async_ld_b32(uint32_t lds_byte_off, const float* gaddr) {
    unsigned long long a = (unsigned long long)(uintptr_t)gaddr;
    asm volatile("global_load_async_to_lds_b32 %0, %1, off"
                 :: "v"(lds_byte_off), "v"(a) : "memory");
}
__device__ __forceinline__ void async_ld_b128(uint32_t lds_byte_off, const float* gaddr) {
    unsigned long long a = (unsigned long long)(uintptr_t)gaddr;
    asm volatile("global_load_async_to_lds_b128 %0, %1, off"
                 :: "v"(lds_byte_off), "v"(a) : "memory");
}
__device__ __forceinline__ void wait_async0() {
    asm volatile("s_wait_asynccnt 0" ::: "memory");
}

// ---------------- kernel 1: per-batch partial sum / sumsq (fixed order => deterministic)
__global__ void __launch_bounds__(THREADS)
reduce_kernel(const float* __restrict__ x, double2* __restrict__ partial) {
    const int b = blockIdx.y, chunk = blockIdx.x, tid = threadIdx.x;
    const float* xb = x + (long)b * T_ + (long)chunk * CHUNK_LEN;  // 16B aligned
    double s = 0.0, s2 = 0.0;
    for (int pv = tid; pv < CHUNK_LEN / 4; pv += THREADS) {
        v4f v = __builtin_nontemporal_load((const v4f*)xb + pv);
        double a0 = (double)v.x, a1 = (double)v.y, a2 = (double)v.z, a3 = (double)v.w;
        s  += a0 + a1 + a2 + a3;
        s2 += a0 * a0 + a1 * a1 + a2 * a2 + a3 * a3;
    }
    __shared__ double rs[THREADS];
    __shared__ double rq[THREADS];
    rs[tid] = s; rq[tid] = s2;
    __syncthreads();
    for (int off = THREADS / 2; off > 0; off >>= 1) {
        if (tid < off) { rs[tid] += rs[tid + off]; rq[tid] += rq[tid + off]; }
        __syncthreads();
    }
    if (tid == 0) partial[b * CHUNKS + chunk] = make_double2(rs[0], rq[0]);
}

// ---------------- kernel 2: fold partials -> per-batch params
// params[b*8 + {0:g, 1:noise_coef, 2:1/k, 3:halfwidth, 4:mode}]
__global__ void params_kernel(const double2* __restrict__ partial,
                              const float* __restrict__ r_gain,
                              const float* __restrict__ r_gval,
                              const float* __restrict__ r_noise,
                              const float* __restrict__ r_namp,
                              const float* __restrict__ r_filt,
                              const float* __restrict__ r_side,
                              const int*   __restrict__ k_raw,
                              float* __restrict__ params) {
    const int b = threadIdx.x;
    if (b >= B_) return;
    double s = 0.0, s2 = 0.0;
    for (int c2 = 0; c2 < CHUNKS; ++c2) {
        double2 p = partial[b * CHUNKS + c2];
        s += p.x; s2 += p.y;
    }
    const double N = (double)T_;
    double var = (s2 - s * s / N) / (N - 1.0);
    if (var < 0.0) var = 0.0;
    const float g = (r_gain[b] < 0.7f) ? (0.7f + 0.6f * r_gval[b]) : 1.0f;
    float stdxi = g * (float)sqrt(var);      // std(g*x) == g*std(x), g > 0
    if (stdxi < 1e-4f) stdxi = 1e-4f;        // clip AFTER gain, matching reference
    const float amp = 0.002f + 0.018f * r_namp[b];
    const float c = (r_noise[b] < 0.5f) ? stdxi * amp : 0.0f;
    const int k = (5 + k_raw[b]) | 1;        // odd in [5,33]
    const float mode = (r_filt[b] < 0.2f) ? ((r_side[b] < 0.5f) ? 1.0f : 2.0f) : 0.0f;
    float* pb = params + b * 8;
    pb[0] = g;
    pb[1] = c;
    pb[2] = 1.0f / (float)k;
    pb[3] = (float)(k / 2);
    pb[4] = mode;
}

// ---------------- kernel 3: fused gain + noise + sliding-window avgpool
__global__ void __launch_bounds__(THREADS)
aug_kernel(const float* __restrict__ x, const float* __restrict__ noise,
           const float* __restrict__ params, float* __restrict__ out) {
    const int b   = blockIdx.y;
    const int tid = threadIdx.x;
    const long tile0 = (long)blockIdx.x * TILE;
    const float* pb = params + b * 8;
    const float g = pb[0], c = pb[1], inv_k = pb[2];
    const int hw   = (int)pb[3];
    const int mode = (int)pb[4];
    const long base = (long)b * T_;

    if (mode == 0) {
        // passthrough batch: pure streaming  out = g*x + c*noise
        for (int e = tid * 4; e < TILE; e += THREADS * 4) {
            const long t = tile0 + e;
            if (t >= T_) break;
            if (t + 4 <= T_) {
                v4f xv = __builtin_nontemporal_load((const v4f*)(x + base + t));
                v4f nv = __builtin_nontemporal_load((const v4f*)(noise + base + t));
                v4f o = g * xv + c * nv;
                __builtin_nontemporal_store(o, (v4f*)(out + base + t));
            } else {
                for (int q = 0; q < 4 && t + q < T_; ++q)
                    out[base + t + q] = g * x[base + t + q] + c * noise[base + t + q];
            }
        }
        return;
    }

    // filtered batch: async-stage raw tiles into LDS (ASYNCcnt path)
    __shared__ __align__(16) float sx [HALO_LEN];
    __shared__ __align__(16) float sn [HALO_LEN];
    __shared__ __align__(16) float sxi[HALO_LEN];
    const uint32_t sx_off = (uint32_t)(uintptr_t)&sx[0];
    const uint32_t sn_off = (uint32_t)(uintptr_t)&sn[0];

    // interior tiles: halo fully inside the batch -> wide b128 async staging
    const bool interior = (tile0 - HW >= 0) && (tile0 + TILE + HW <= T_);

    if (interior) {
        const float* gx = x + base + tile0 - HW;      // 64B aligned
        const float* gn = noise + base + tile0 - HW;
        for (int pv = tid; pv < HALO_LEN / 4; pv += THREADS) {
            async_ld_b128(sx_off + 16u * (uint32_t)pv, gx + 4 * pv);
            async_ld_b128(sn_off + 16u * (uint32_t)pv, gn + 4 * pv);
        }
    } else {
        for (int p = tid; p < HALO_LEN; p += THREADS) {
            long t = tile0 - HW + p;
            long tc = t < 0 ? 0 : (t >= T_ ? (long)(T_ - 1) : t);   // clamp; masked below
            async_ld_b32(sx_off + 4u * (uint32_t)p, x + base + tc);
            async_ld_b32(sn_off + 4u * (uint32_t)p, noise + base + tc);
        }
    }
    wait_async0();
    __syncthreads();

    // form xi (conv zero-padding applied at batch boundaries)
    if (interior) {
        for (int pv = tid; pv < HALO_LEN / 4; pv += THREADS) {
            v4f a  = ((const v4f*)sx)[pv];
            v4f nn = ((const v4f*)sn)[pv];
            ((v4f*)sxi)[pv] = g * a + c * nn;
        }
    } else {
        for (int p = tid; p < HALO_LEN; p += THREADS) {
            long t = tile0 - HW + p;
            sxi[p] = (t >= 0 && t < T_) ? (g * sx[p] + c * sn[p]) : 0.0f;
        }
    }
    __syncthreads();

    const int lbase = HW + tid * ELEMS;
    float w = 0.0f;
    for (int j = -hw; j <= hw; ++j) w += sxi[lbase + j];

    float res[ELEMS];
    for (int e = 0; e < ELEMS; ++e) {
        const float xc = sxi[lbase + e];
        const float lo = w * inv_k;
        res[e] = (mode == 1) ? lo : (xc - lo);
        if (e + 1 < ELEMS)
            w += sxi[lbase + e + 1 + hw] - sxi[lbase + e - hw];  // slide window
    }

    const long g0 = tile0 + (long)tid * ELEMS;
    if (g0 + ELEMS <= T_) {
        v4f lo4 = { res[0], res[1], res[2], res[3] };
        v4f hi4 = { res[4], res[5], res[6], res[7] };
        __builtin_nontemporal_store(lo4, (v4f*)(out + base + g0));
        __builtin_nontemporal_store(hi4, (v4f*)(out + base + g0 + 4));
    } else {
        for (int e = 0; e < ELEMS; ++e)
            if (g0 + e < T_) out[base + g0 + e] = res[e];
    }
}

extern "C" void kernel_launch(void* const* d_in, const int* in_sizes, int n_in,
                              void* d_out, int out_size, void* d_ws, size_t ws_size,
                              hipStream_t stream) {
    const float* x       = (const float*)d_in[0];
    const float* noise   = (const float*)d_in[1];
    const float* r_gain  = (const float*)d_in[2];
    const float* r_gval  = (const float*)d_in[3];
    const float* r_noise = (const float*)d_in[4];
    const float* r_namp  = (const float*)d_in[5];
    const float* r_filt  = (const float*)d_in[6];
    const float* r_side  = (const float*)d_in[7];
    const int*   k_raw   = (const int*)d_in[8];
    float* out = (float*)d_out;

    double2* partial = (double2*)d_ws;                                   // 512 * 16 B
    float*   params  = (float*)((char*)d_ws + B_ * CHUNKS * sizeof(double2)); // 64*8 f32

    reduce_kernel<<<dim3(CHUNKS, B_), THREADS, 0, stream>>>(x, partial);
    params_kernel<<<1, 64, 0, stream>>>(partial, r_gain, r_gval, r_noise, r_namp,
                                        r_filt, r_side, k_raw, params);
    aug_kernel<<<dim3((T_ + TILE - 1) / TILE, B_), THREADS, 0, stream>>>(x, noise, params, out);
}